// SAGE_90366111908390
// MI455X (gfx1250) — compile-verified
//
#include <hip/hip_runtime.h>

typedef __attribute__((ext_vector_type(2))) float v2f;
typedef __attribute__((ext_vector_type(8))) float v8f;

#define NFEAT 128

// ---------------------------------------------------------------- utilities
__global__ void zero_f32(float* __restrict__ p, size_t count) {
  size_t i = (size_t)blockIdx.x * blockDim.x + threadIdx.x;
  size_t stride = (size_t)gridDim.x * blockDim.x;
  for (; i < count; i += stride) p[i] = 0.0f;
}

// one thread per edge: deg[dst] += 1
__global__ void deg_kernel(const int* __restrict__ dst, float* __restrict__ deg,
                           int nE) {
  int e = blockIdx.x * blockDim.x + threadIdx.x;
  if (e < nE) atomicAdd(&deg[dst[e]], 1.0f);
}

// one wave32 per edge: gather full 128-float row of feat[src], atomic-add
// into agg[dst]. float4 per lane => 512B coalesced gather + coalesced atomics.
// Working set (agg = 25.6MB) lives in the 192MB L2, so atomics stay on-chip.
__global__ void scatter_kernel(const float* __restrict__ feat,
                               const int* __restrict__ src,
                               const int* __restrict__ dst,
                               float* __restrict__ agg, int nE) {
  int gw = (int)(((size_t)blockIdx.x * blockDim.x + threadIdx.x) >> 5);
  int lane = threadIdx.x & 31;
  if (gw >= nE) return;
  int s = src[gw];
  int d = dst[gw];
  float4 v = ((const float4*)(feat + (size_t)s * NFEAT))[lane];
  float* o = agg + (size_t)d * NFEAT + (size_t)lane * 4;
  atomicAdd(o + 0, v.x);
  atomicAdd(o + 1, v.y);
  atomicAdd(o + 2, v.z);
  atomicAdd(o + 3, v.w);
}

// ------------------------------------------------- fused SAGE layer (WMMA)
// out[i,:] = act( (agg[i,:]/max(deg[i],1)) @ Wl + xin[i,:] @ Wr + bias )
// One wave computes one 16x16 output tile with V_WMMA_F32_16X16X4_F32
// (exact fp32 semantics). Block = (DOUT/16) waves sharing 16 node rows.
template <int DOUT, bool RELU>
__global__ void sage_gemm(const float* __restrict__ agg,
                          const float* __restrict__ deg,
                          const float* __restrict__ xin,
                          const float* __restrict__ Wl,
                          const float* __restrict__ Wr,
                          const float* __restrict__ bias,
                          float* __restrict__ out, int n) {
  const int lane = threadIdx.x & 31;
  const int wv   = threadIdx.x >> 5;
  const int col0 = wv * 16;
  const int row0 = blockIdx.x * 16;
  const int m  = lane & 15;   // A row (M) for lanes, also B/C column (N)
  const int hi = lane >> 4;   // K half-select / C row-half select
  const int kb = hi * 2;

  int rowA = row0 + m;
  if (rowA > n - 1) rowA = n - 1;  // clamp loads; keeps EXEC all-ones for WMMA

  const float dg  = deg[rowA];
  const float inv = 1.0f / fmaxf(dg, 1.0f);

  const float* aggRow = agg + (size_t)rowA * NFEAT;
  const float* xRow   = xin + (size_t)rowA * NFEAT;

  v8f acc = {};

  // mean-aggregated neighbors @ Wl
#pragma unroll 8
  for (int k0 = 0; k0 < NFEAT; k0 += 4) {
    const float2 av = *(const float2*)(aggRow + k0 + kb);
    v2f a, b;
    a.x = av.x * inv;
    a.y = av.y * inv;
    b.x = Wl[(size_t)(k0 + kb) * DOUT + col0 + m];
    b.y = Wl[(size_t)(k0 + kb + 1) * DOUT + col0 + m];
    acc = __builtin_amdgcn_wmma_f32_16x16x4_f32(false, a, false, b, (short)0,
                                                acc, false, false);
  }
  // self features @ Wr (accumulate into same C)
#pragma unroll 8
  for (int k0 = 0; k0 < NFEAT; k0 += 4) {
    const float2 av = *(const float2*)(xRow + k0 + kb);
    v2f a, b;
    a.x = av.x;
    a.y = av.y;
    b.x = Wr[(size_t)(k0 + kb) * DOUT + col0 + m];
    b.y = Wr[(size_t)(k0 + kb + 1) * DOUT + col0 + m];
    acc = __builtin_amdgcn_wmma_f32_16x16x4_f32(false, a, false, b, (short)0,
                                                acc, false, false);
  }

  const float bv = bias[col0 + m];
#pragma unroll
  for (int r = 0; r < 8; ++r) {
    int row = row0 + r + hi * 8;  // C layout: VGPR r -> M = r + 8*hi
    if (row < n) {
      float v = acc[r] + bv;
      if (RELU) v = fmaxf(v, 0.0f);
      out[(size_t)row * DOUT + col0 + m] = v;
    }
  }
}

// ----------------------------------------------------------------- launcher
extern "C" void kernel_launch(void* const* d_in, const int* in_sizes, int n_in,
                              void* d_out, int out_size, void* d_ws,
                              size_t ws_size, hipStream_t stream) {
  const float* x    = (const float*)d_in[0];
  const int*   eix  = (const int*)d_in[1];
  const float* Wl0  = (const float*)d_in[2];
  const float* Wr0  = (const float*)d_in[3];
  const float* b0   = (const float*)d_in[4];
  const float* Wl1  = (const float*)d_in[5];
  const float* Wr1  = (const float*)d_in[6];
  const float* b1   = (const float*)d_in[7];
  const float* Wl2  = (const float*)d_in[8];
  const float* Wr2  = (const float*)d_in[9];
  const float* b2   = (const float*)d_in[10];

  const int n  = in_sizes[0] / NFEAT;  // 50000 nodes
  const int nE = in_sizes[1] / 2;      // 800000 edges
  const int* src = eix;
  const int* dst = eix + nE;

  // workspace layout: deg (padded) | agg | h1 | h2   (~77 MB)
  float* deg = (float*)d_ws;
  size_t degPad = ((size_t)n + 511) & ~(size_t)511;
  float* agg = deg + degPad;
  float* h1  = agg + (size_t)n * NFEAT;
  float* h2  = h1 + (size_t)n * NFEAT;

  const int tiles      = (n + 15) / 16;
  const int edgeBlocks = (nE * 32 + 255) / 256;  // one wave per edge
  const int degBlocks  = (nE + 255) / 256;

  // --- degree (once) + layer 0 aggregation
  zero_f32<<<2048, 256, 0, stream>>>(deg, degPad + (size_t)n * NFEAT);
  deg_kernel<<<degBlocks, 256, 0, stream>>>(dst, deg, nE);
  scatter_kernel<<<edgeBlocks, 256, 0, stream>>>(x, src, dst, agg, nE);
  sage_gemm<128, true><<<tiles, 256, 0, stream>>>(agg, deg, x, Wl0, Wr0, b0,
                                                  h1, n);
  // --- layer 1
  zero_f32<<<2048, 256, 0, stream>>>(agg, (size_t)n * NFEAT);
  scatter_kernel<<<edgeBlocks, 256, 0, stream>>>(h1, src, dst, agg, nE);
  sage_gemm<128, true><<<tiles, 256, 0, stream>>>(agg, deg, h1, Wl1, Wr1, b1,
                                                  h2, n);
  // --- layer 2 (n_hid -> n_class=64, no ReLU)
  zero_f32<<<2048, 256, 0, stream>>>(agg, (size_t)n * NFEAT);
  scatter_kernel<<<edgeBlocks, 256, 0, stream>>>(h2, src, dst, agg, nE);
  sage_gemm<64, false><<<tiles, 128, 0, stream>>>(agg, deg, h2, Wl2, Wr2, b2,
                                                  (float*)d_out, n);
}